// Sampler_28853590294563
// MI455X (gfx1250) — compile-verified
//
#include <hip/hip_runtime.h>
#include <stdint.h>

#define THREADS 256
#define CHUNK   5120               // floats per staged chunk (20 KB)
#define VPT     (CHUNK / (THREADS * 4))   // float4 async issues per thread = 5
#define BINS    2048

// ---------- helpers ----------

// monotone float -> uint key: larger float <=> larger key
static __device__ __forceinline__ uint32_t fkey(float v) {
    uint32_t u = __float_as_uint(v);
    uint32_t m = (uint32_t)((int32_t)u >> 31) | 0x80000000u;
    return u ^ m;
}

// deterministic uniform in (0,1) from (row, col) — splitmix64-style hash
static __device__ __forceinline__ float uni01(uint32_t row, uint32_t col) {
    uint64_t x = ((uint64_t)row << 32) | (uint64_t)col;
    x ^= 0x2545F4914F6CDD1DULL;            // fixed seed salt
    x += 0x9E3779B97F4A7C15ULL;
    x = (x ^ (x >> 30)) * 0xBF58476D1CE4E5B9ULL;
    x = (x ^ (x >> 27)) * 0x94D049BB133111EBULL;
    x ^= x >> 31;
    uint32_t m = (uint32_t)(x >> 40);       // 24 bits
    return ((float)m + 0.5f) * (1.0f / 16777216.0f);
}

// Stream one row through LDS with double-buffered CDNA5 async copies.
// f(value, global_index) is called for every element of the row.
template <typename F>
static __device__ __forceinline__ void stream_row(const float* __restrict__ row,
                                                  int Vn, float* st0, float* st1,
                                                  F&& f) {
    const int tid = threadIdx.x;
    const int nchunk = Vn / CHUNK;
    float*   bufs[2]  = { st0, st1 };
    uint32_t loffs[2] = { (uint32_t)(uintptr_t)st0, (uint32_t)(uintptr_t)st1 };

    auto issue = [&](int chunk, int b) {
        const float* g = row + (size_t)chunk * CHUNK;
#pragma unroll
        for (int j = 0; j < VPT; ++j) {
            int e = (j * THREADS + tid) * 4;       // float index within chunk
            const float* gp = g + e;
            uint32_t lo = loffs[b] + (uint32_t)(e * 4);
            // CDNA5 async copy: 16B global -> LDS per lane, tracked by ASYNCcnt
            asm volatile("global_load_async_to_lds_b128 %0, %1, off"
                         :: "v"(lo), "v"(gp) : "memory");
        }
    };

    if (nchunk > 0) issue(0, 0);
    for (int c = 0; c < nchunk; ++c) {
        int b = c & 1;
        if (c + 1 < nchunk) {
            issue(c + 1, b ^ 1);                   // keep next chunk in flight
            asm volatile("s_wait_asynccnt 5" ::: "memory");  // retire chunk c
        } else {
            asm volatile("s_wait_asynccnt 0" ::: "memory");
        }
        __syncthreads();                           // publish chunk c to all waves
        const float* sb = bufs[b];
        const int base = c * CHUNK;
#pragma unroll 4
        for (int e = tid; e < CHUNK; e += THREADS) f(sb[e], base + e);
        __syncthreads();                           // buffer safe for reuse
    }
    // generic tail (V % CHUNK == 0 for this problem, but stay safe)
    for (int i = nchunk * CHUNK + tid; i < Vn; i += THREADS) f(row[i], i);
}

// ---------- kernel: one workgroup per row ----------

__global__ void __launch_bounds__(THREADS)
topk_sample_kernel(const float* __restrict__ logits,
                   const float* __restrict__ temps,
                   const int* __restrict__ topk_p,
                   float* __restrict__ out, int Vn) {
    __shared__ float    stage0[CHUNK];
    __shared__ float    stage1[CHUNK];
    __shared__ uint32_t hist[BINS];
    __shared__ uint32_t scanArr[THREADS];
    __shared__ float    bestScore[THREADS];
    __shared__ uint32_t bestIdx[THREADS];
    __shared__ uint32_t s_selBin, s_kRem;

    const int row = blockIdx.x;
    const int tid = threadIdx.x;
    const float* rptr = logits + (size_t)row * (size_t)Vn;

    int k = topk_p[0];
    if (k < 1) k = 1;
    if (k > Vn) k = Vn;

    // ---- 3-round radix select for the k-th largest 32-bit key ----
    const int shifts[3] = { 21, 10, 0 };
    const int fbits[3]  = { 11, 11, 10 };
    uint32_t prefix = 0;
    int      pbits  = 0;
    uint32_t kRem   = (uint32_t)k;

    for (int r = 0; r < 3; ++r) {
        const int shift = shifts[r];
        const uint32_t nbm = (1u << fbits[r]) - 1u;

        for (int i = tid; i < BINS; i += THREADS) hist[i] = 0u;
        if (tid == 0) { s_selBin = 0u; s_kRem = 1u; }
        __syncthreads();

        const uint32_t pv = prefix;
        const int      pb = pbits;
        stream_row(rptr, Vn, stage0, stage1, [&](float v, int idx) {
            (void)idx;
            uint32_t key = fkey(v);
            bool m = (pb == 0) || ((key >> (32 - pb)) == pv);
            if (m) atomicAdd(&hist[(key >> shift) & nbm], 1u);
        });
        __syncthreads();

        // parallel suffix scan over 256 segments of 8 bins each
        uint32_t seg = 0u;
        const int BPT = BINS / THREADS;            // 8
#pragma unroll
        for (int j = 0; j < BPT; ++j) seg += hist[tid * BPT + j];
        scanArr[tid] = seg;
        __syncthreads();
#pragma unroll
        for (int off = 1; off < THREADS; off <<= 1) {
            uint32_t v2 = (tid + off < THREADS) ? scanArr[tid + off] : 0u;
            __syncthreads();
            scanArr[tid] += v2;
            __syncthreads();
        }
        uint32_t above = scanArr[tid] - seg;       // count strictly above my segment
        if (above < kRem && above + seg >= kRem) { // unique owner of the k-th
            uint32_t acc = above;
            for (int b = tid * BPT + BPT - 1; b >= tid * BPT; --b) {
                uint32_t c = hist[b];
                if (acc + c >= kRem) { s_selBin = (uint32_t)b; s_kRem = kRem - acc; break; }
                acc += c;
            }
        }
        __syncthreads();
        prefix = (prefix << fbits[r]) | s_selBin;
        pbits += fbits[r];
        kRem = s_kRem;
        __syncthreads();
    }
    const uint32_t T = prefix;                     // exact key of k-th largest

    // ---- Gumbel-max sampling over candidates (key >= T) ----
    const float invT = 1.0f / temps[row];
    float    bScore = -INFINITY;
    uint32_t bIdx   = 0xFFFFFFFFu;
    stream_row(rptr, Vn, stage0, stage1, [&](float v, int idx) {
        uint32_t key = fkey(v);
        if (key >= T) {
            float u = uni01((uint32_t)row, (uint32_t)idx);
            float g = -__logf(-__logf(u));
            float sc = v * invT + g;
            if (sc > bScore || (sc == bScore && (uint32_t)idx < bIdx)) {
                bScore = sc; bIdx = (uint32_t)idx;
            }
        }
    });
    bestScore[tid] = bScore;
    bestIdx[tid]   = bIdx;
    __syncthreads();
#pragma unroll
    for (int off = THREADS / 2; off > 0; off >>= 1) {
        if (tid < off) {
            float s2 = bestScore[tid + off];
            uint32_t i2 = bestIdx[tid + off];
            if (s2 > bestScore[tid] || (s2 == bestScore[tid] && i2 < bestIdx[tid])) {
                bestScore[tid] = s2; bestIdx[tid] = i2;
            }
        }
        __syncthreads();
    }
    if (tid == 0) out[row] = (float)bestIdx[0];
}

// ---------- host launch ----------

extern "C" void kernel_launch(void* const* d_in, const int* in_sizes, int n_in,
                              void* d_out, int out_size, void* d_ws, size_t ws_size,
                              hipStream_t stream) {
    (void)n_in; (void)d_ws; (void)ws_size; (void)out_size;
    const float* logits = (const float*)d_in[0];
    const float* temps  = (const float*)d_in[1];
    const int*   topk   = (const int*)d_in[2];
    float*       out    = (float*)d_out;

    const int B = in_sizes[1];
    const int V = in_sizes[0] / B;

    topk_sample_kernel<<<dim3(B), dim3(THREADS), 0, stream>>>(logits, temps, topk, out, V);
}